// DataFKImageNetwork_17686675325043
// MI455X (gfx1250) — compile-verified
//
#include <hip/hip_runtime.h>
#include <math.h>

// ---------------- problem constants ----------------
#define NB   4
#define NT   2048
#define NXR  128
#define NTF  8192      // 4*NT
#define NXF  512       // 2*ceil(4*NX/2)
#define NF   4097      // NTF/2+1
#define FS_      40000000.0f
#define PITCH_   0.0003f
#define C_       1540.0f
#define DF_      (FS_/ (float)NTF)        // 4882.8125
#define EPS_     1.1920929e-07f
#define SIN_A    0.09983341664682815f
#define COS_A    0.99500416527802577f
#define TWO_PI_  6.28318530717958647692f

typedef __attribute__((ext_vector_type(2))) float v2f;
typedef __attribute__((ext_vector_type(8))) float v8f;

__device__ __forceinline__ unsigned bitrev(unsigned x, int bits) {
    return __brev(x) >> (32 - bits);
}

// ---------------- generic LDS radix-2 FFTs ----------------
// DIF: natural input -> bit-reversed output. sign = -1 fwd, +1 inv (unscaled).
__device__ void fft_dif(float* re, float* im, int log2N, float sign) {
    const int half = 1 << (log2N - 1);
    for (int ls = log2N - 1; ls >= 0; --ls) {
        const int span = 1 << ls;
        for (int j = threadIdx.x; j < half; j += blockDim.x) {
            int pos = j & (span - 1);
            int blk = j >> ls;
            int i0  = (blk << (ls + 1)) + pos;
            int i1  = i0 + span;
            float ur = re[i0], ui = im[i0];
            float vr = re[i1], vi = im[i1];
            re[i0] = ur + vr; im[i0] = ui + vi;
            float ang = sign * TWO_PI_ * (float)pos / (float)(span << 1);
            float c, s; __sincosf(ang, &s, &c);
            float dr = ur - vr, di = ui - vi;
            re[i1] = dr * c - di * s;
            im[i1] = dr * s + di * c;
        }
        __syncthreads();
    }
}

// DIT: bit-reversed input -> natural output. sign = +1 inv (unscaled).
__device__ void fft_dit(float* re, float* im, int log2N, float sign) {
    const int half = 1 << (log2N - 1);
    for (int ls = 0; ls < log2N; ++ls) {
        const int span = 1 << ls;
        for (int j = threadIdx.x; j < half; j += blockDim.x) {
            int pos = j & (span - 1);
            int blk = j >> ls;
            int i0  = (blk << (ls + 1)) + pos;
            int i1  = i0 + span;
            float ang = sign * TWO_PI_ * (float)pos / (float)(span << 1);
            float c, s; __sincosf(ang, &s, &c);
            float vr = re[i1] * c - im[i1] * s;
            float vi = re[i1] * s + im[i1] * c;
            float ur = re[i0], ui = im[i0];
            re[i0] = ur + vr; im[i0] = ui + vi;
            re[i1] = ur - vr; im[i1] = ui - vi;
        }
        __syncthreads();
    }
}

// ---------------- WMMA-accelerated four-step 8192-pt FFT ----------------
// N = 16 * 512.  n = n1*512 + n2, k = k1 + 16*k2.
// Stage A: A[k1,n2] = sum_n1 x[n1*512+n2] * exp(sign*i*2pi*k1*n1/16)   (WMMA f32 16x16x4)
// Twiddle: A[k1,n2] *= exp(sign*i*2pi*k1*n2/8192)
// Stage B: 16 row-wise 512-pt DIF FFTs (natural -> bitrev)
// Result element k lands in re and im at index (k&15)*512 + bitrev(k>>4, 9).
__device__ __forceinline__ v8f wmma4(v2f a, v2f b, v8f c) {
    return __builtin_amdgcn_wmma_f32_16x16x4_f32(false, a, false, b, (short)0, c, false, false);
}

__device__ void fft8192_fourstep(float* re, float* im, float sign) {
    const int tid   = threadIdx.x;           // 512 threads = 16 waves (wave32)
    const int wave  = tid >> 5;
    const int lane  = tid & 31;
    const int mrow  = lane & 15;             // A row / B-D column within tile
    const int khalf = (lane < 16) ? 0 : 2;   // K split per ISA 16x4 A layout

    // ---- Stage A: each wave owns 2 disjoint 16-column tiles (32 tiles total) ----
    for (int tile = wave; tile < 32; tile += 16) {
        const int n2 = tile * 16 + mrow;     // this lane's B/D column
        v8f rr = {}; v8f ri = {}; v8f ir = {}; v8f ii = {};
        #pragma unroll
        for (int c4 = 0; c4 < 4; ++c4) {
            int k0 = 4 * c4 + khalf;
            int k1 = k0 + 1;
            // A lanes: F16[mrow, k0], F16[mrow, k1]  (F = exp(sign*i*2pi*m*n/16))
            float a0 = sign * TWO_PI_ * (float)((mrow * k0) & 15) * (1.0f / 16.0f);
            float a1 = sign * TWO_PI_ * (float)((mrow * k1) & 15) * (1.0f / 16.0f);
            float c0, s0, c1, s1;
            __sincosf(a0, &s0, &c0);
            __sincosf(a1, &s1, &c1);
            v2f ar = { c0, c1 };
            v2f ai = { s0, s1 };
            // B lanes: X[k0, n2], X[k1, n2]
            v2f br = { re[k0 * 512 + n2], re[k1 * 512 + n2] };
            v2f bi = { im[k0 * 512 + n2], im[k1 * 512 + n2] };
            rr = wmma4(ar, br, rr);
            ii = wmma4(ai, bi, ii);
            ri = wmma4(ar, bi, ri);
            ir = wmma4(ai, br, ir);
        }
        // D layout: element (row = v + (lane<16?0:8), col = lane&15); col == our n2
        #pragma unroll
        for (int v = 0; v < 8; ++v) {
            int krow = v + ((lane < 16) ? 0 : 8);
            float cr = rr[v] - ii[v];
            float ci = ri[v] + ir[v];
            float ang = sign * TWO_PI_ * (float)(krow * n2) * (1.0f / 8192.0f);
            float c, s; __sincosf(ang, &s, &c);
            re[krow * 512 + n2] = cr * c - ci * s;
            im[krow * 512 + n2] = cr * s + ci * c;
        }
    }
    __syncthreads();

    // ---- Stage B: 16 rows x 512-pt DIF, block-cooperative ----
    for (int ls = 8; ls >= 0; --ls) {
        const int span = 1 << ls;
        for (int j = tid; j < 16 * 256; j += 512) {
            int row = j >> 8;
            int jj  = j & 255;
            int pos = jj & (span - 1);
            int blk = jj >> ls;
            int i0  = row * 512 + (blk << (ls + 1)) + pos;
            int i1  = i0 + span;
            float ur = re[i0], ui = im[i0];
            float vr = re[i1], vi = im[i1];
            re[i0] = ur + vr; im[i0] = ui + vi;
            float ang = sign * TWO_PI_ * (float)pos / (float)(span << 1);
            float c, s; __sincosf(ang, &s, &c);
            float dr = ur - vr, di = ui - vi;
            re[i1] = dr * c - di * s;
            im[i1] = dr * s + di * c;
        }
        __syncthreads();
    }
}

__device__ __forceinline__ int fs_idx(int k) {
    return (k & 15) * 512 + (int)bitrev((unsigned)(k >> 4), 9);
}

// ---------------- Kernel 1: per-column 8192-pt FFT + tx-delay phase ----------------
__global__ __launch_bounds__(512) void k1_timefft(const float* __restrict__ data,
                                                  float2* __restrict__ ola) {
    __shared__ float re[NTF];
    __shared__ float im[NTF];
    const int b = blockIdx.x >> 7;
    const int c = blockIdx.x & 127;
    const float* col = data + (size_t)b * NT * NXR + c;
    for (int i = threadIdx.x; i < NTF; i += 512) {
        re[i] = (i < NT) ? col[(size_t)i * NXR] : 0.0f;
        im[i] = 0.0f;
    }
    __syncthreads();
    fft8192_fourstep(re, im, -1.0f);
    // t_delay = sin_a * (0 - c) * PITCH/C  (TX_ANGLE > 0, T0 = 0)
    const float td = -SIN_A * (float)c * (PITCH_ / C_);
    for (int r = threadIdx.x; r < NF; r += 512) {
        int idx = fs_idx(r);
        float xr = re[idx], xi = im[idx];
        float ang = -TWO_PI_ * td * ((float)r * DF_);
        float cc, ss; __sincosf(ang, &ss, &cc);
        ola[((size_t)b * NF + r) * NXR + c] = make_float2(xr * cc - xi * ss,
                                                          xr * ss + xi * cc);
    }
}

// ---------------- Kernel 2: per-row 512-pt x-FFT + evanescent mask ----------------
__global__ __launch_bounds__(256) void k2_xfft(const float2* __restrict__ ola,
                                               float2* __restrict__ olb) {
    __shared__ float re[NXF];
    __shared__ float im[NXF];
    const int b = blockIdx.x / NF;
    const int r = blockIdx.x - b * NF;
    const float2* row = ola + ((size_t)b * NF + r) * NXR;
    for (int i = threadIdx.x; i < NXF; i += 256) {
        if (i < NXR) { float2 v = row[i]; re[i] = v.x; im[i] = v.y; }
        else         { re[i] = 0.0f;     im[i] = 0.0f; }
    }
    __syncthreads();
    fft_dif(re, im, 9, -1.0f);              // spectrum in bitrev order
    const float f = (float)r * DF_;
    float2* orow = olb + ((size_t)b * NF + r) * NXF;
    for (int p = threadIdx.x; p < NXF; p += 256) {
        int k   = (int)bitrev((unsigned)p, 9);
        int kxi = (k <= 256) ? k : k - 512;
        float kx = (float)kxi * (1.0f / (PITCH_ * (float)NXF));
        bool evan = (fabsf(f) / (fabsf(kx) + EPS_)) < C_;
        orow[k] = evan ? make_float2(0.0f, 0.0f) : make_float2(re[p], im[p]);
    }
}

// ---- Kernel 3: Stolt remap + symmetric spectrum + 8192 IFFT + kx phase ----
__global__ __launch_bounds__(512) void k3_stolt(const float2* __restrict__ olb,
                                                float2* __restrict__ td) {
    __shared__ float re[NTF];
    __shared__ float im[NTF];
    const int b  = blockIdx.x >> 9;
    const int k  = blockIdx.x & 511;
    const int kp = (NXF - k) & 511;          // mirror column: kx' = -kx (kx^2 equal)
    const int kxi = (k <= 256) ? k : k - 512;
    const float kx = (float)kxi * (1.0f / (PITCH_ * (float)NXF));

    const float denom = 1.0f + COS_A + SIN_A * SIN_A;
    const float v_erm = C_ / sqrtf(denom);
    const float beta  = powf(1.0f + COS_A, 1.5f) / denom;

    const float2* colk = olb + (size_t)b * NF * NXF + k;
    const float2* colp = olb + (size_t)b * NF * NXF + kp;

    for (int i = threadIdx.x; i < NF; i += 512) {
        float f   = (float)i * DF_;
        float kz  = 2.0f * f / (beta * C_);
        float fkz = v_erm * sqrtf(kx * kx + kz * kz);
        float iq  = fkz * (1.0f / DF_);
        float vr = 0.0f, vi = 0.0f, wr = 0.0f, wi = 0.0f;
        if (iq < (float)(NF - 2)) {
            int   fl = (int)floorf(iq);
            float lw = iq - (float)fl;
            float2 a0 = colk[(size_t)fl * NXF];
            float2 a1 = colk[(size_t)(fl + 1) * NXF];
            vr = a0.x * (1.0f - lw) + a1.x * lw;
            vi = a0.y * (1.0f - lw) + a1.y * lw;
            float2 b0 = colp[(size_t)fl * NXF];
            float2 b1 = colp[(size_t)(fl + 1) * NXF];
            wr = b0.x * (1.0f - lw) + b1.x * lw;
            wi = b0.y * (1.0f - lw) + b1.y * lw;
        }
        float fac = f / (fkz + EPS_);
        vr *= fac; vi *= fac; wr *= fac; wi *= fac;
        if (i == 0) { vr = 0.0f; vi = 0.0f; }
        re[i] = vr; im[i] = vi;
        if (i >= 1 && i <= NTF / 2 - 1) {    // full[8192-i,k] = conj(interp_k'[i])
            re[NTF - i] = wr;
            im[NTF - i] = -wi;
        }
    }
    __syncthreads();
    fft8192_fourstep(re, im, +1.0f);
    const float gamma = SIN_A / (2.0f - COS_A);
    for (int t = threadIdx.x; t < NT; t += 512) {
        int idx = fs_idx(t);
        float xr = re[idx] * (1.0f / (float)NTF);
        float xi = im[idx] * (1.0f / (float)NTF);
        float dx  = -gamma * ((float)t / FS_) * C_ * 0.5f;
        float ang = -TWO_PI_ * kx * dx;
        float cc, ss; __sincosf(ang, &ss, &cc);
        td[((size_t)b * NT + t) * NXF + k] = make_float2(xr * cc - xi * ss,
                                                         xr * ss + xi * cc);
    }
}

// ---------------- Kernel 4: per-row 512-pt x-IFFT, keep real of cols 0..127 ----
__global__ __launch_bounds__(256) void k4_xifft(const float2* __restrict__ td,
                                                float* __restrict__ mig) {
    __shared__ float re[NXF];
    __shared__ float im[NXF];
    const int b = blockIdx.x >> 11;
    const int t = blockIdx.x & 2047;
    const float2* row = td + ((size_t)b * NT + t) * NXF;
    for (int i = threadIdx.x; i < NXF; i += 256) {
        float2 v = row[i]; re[i] = v.x; im[i] = v.y;
    }
    __syncthreads();
    fft_dif(re, im, 9, +1.0f);               // inverse, bitrev output
    for (int c = threadIdx.x; c < NXR; c += 256) {
        mig[((size_t)b * NT + t) * NXR + c] =
            re[bitrev((unsigned)c, 9)] * (1.0f / (float)NXF);
    }
}

// ---------------- Kernel 5: Hilbert envelope + log + per-column max ----------------
__global__ __launch_bounds__(256) void k5_hilbert(const float* __restrict__ mig,
                                                  float* __restrict__ logv,
                                                  float* __restrict__ colmax) {
    __shared__ float re[NT];
    __shared__ float im[NT];
    __shared__ float smax[256];
    const int b = blockIdx.x >> 7;
    const int c = blockIdx.x & 127;
    const float* col = mig + (size_t)b * NT * NXR + c;
    for (int i = threadIdx.x; i < NT; i += 256) {
        re[i] = col[(size_t)i * NXR];
        im[i] = 0.0f;
    }
    __syncthreads();
    fft_dif(re, im, 11, -1.0f);              // bitrev spectrum
    for (int p = threadIdx.x; p < NT; p += 256) {
        int k = (int)bitrev((unsigned)p, 11);
        float h = (k == 0 || k == NT / 2) ? 1.0f : ((k < NT / 2) ? 2.0f : 0.0f);
        re[p] *= h; im[p] *= h;
    }
    __syncthreads();
    fft_dit(re, im, 11, +1.0f);              // natural-order analytic signal
    float lmax = -3.0e38f;
    for (int i = threadIdx.x; i < NT; i += 256) {
        float ar = re[i] * (1.0f / (float)NT);
        float ai = im[i] * (1.0f / (float)NT);
        float lv = 20.0f * log10f(sqrtf(ar * ar + ai * ai));
        logv[((size_t)b * NT + i) * NXR + c] = lv;
        lmax = fmaxf(lmax, lv);
    }
    smax[threadIdx.x] = lmax;
    __syncthreads();
    for (int s = 128; s > 0; s >>= 1) {
        if (threadIdx.x < s) smax[threadIdx.x] = fmaxf(smax[threadIdx.x], smax[threadIdx.x + s]);
        __syncthreads();
    }
    if (threadIdx.x == 0) colmax[b * NXR + c] = smax[0];
}

// ---------------- Kernel 6: per-batch max ----------------
__global__ __launch_bounds__(128) void k6_bmax(const float* __restrict__ colmax,
                                               float* __restrict__ bmax) {
    __shared__ float s[128];
    s[threadIdx.x] = colmax[blockIdx.x * NXR + threadIdx.x];
    __syncthreads();
    for (int st = 64; st > 0; st >>= 1) {
        if (threadIdx.x < st) s[threadIdx.x] = fmaxf(s[threadIdx.x], s[threadIdx.x + st]);
        __syncthreads();
    }
    if (threadIdx.x == 0) bmax[blockIdx.x] = s[0];
}

// ---------------- Kernel 7: clip + normalize ----------------
__global__ __launch_bounds__(256) void k7_norm(const float* __restrict__ logv,
                                               const float* __restrict__ bmax,
                                               float* __restrict__ out) {
    int i = blockIdx.x * 256 + threadIdx.x;
    int b = i / (NT * NXR);
    float v = logv[i] - bmax[b];
    v = fmaxf(v, -70.0f);
    out[i] = (v + 70.0f) * (1.0f / 70.0f);
}

// ---------------- launch ----------------
extern "C" void kernel_launch(void* const* d_in, const int* in_sizes, int n_in,
                              void* d_out, int out_size, void* d_ws, size_t ws_size,
                              hipStream_t stream) {
    (void)in_sizes; (void)n_in; (void)out_size; (void)ws_size;
    const float* data = (const float*)d_in[0];
    float* out = (float*)d_out;

    char* ws = (char*)d_ws;
    size_t off = 0;
    auto alloc = [&](size_t bytes) -> void* {
        void* p = ws + off;
        off += (bytes + 255) & ~(size_t)255;
        return p;
    };
    float2* ola  = (float2*)alloc((size_t)NB * NF * NXR * sizeof(float2)); // 16.8 MB
    float2* olb  = (float2*)alloc((size_t)NB * NF * NXF * sizeof(float2)); // 67.1 MB
    float2* td   = (float2*)alloc((size_t)NB * NT * NXF * sizeof(float2)); // 33.5 MB
    float*  mig  = (float*) alloc((size_t)NB * NT * NXR * sizeof(float));  //  4.2 MB
    float*  logv = (float*) alloc((size_t)NB * NT * NXR * sizeof(float));  //  4.2 MB
    float*  cmax = (float*) alloc((size_t)NB * NXR * sizeof(float));
    float*  bmax = (float*) alloc((size_t)NB * sizeof(float));

    k1_timefft<<<NB * NXR, 512, 0, stream>>>(data, ola);
    k2_xfft  <<<NB * NF, 256, 0, stream>>>(ola, olb);
    k3_stolt <<<NB * NXF, 512, 0, stream>>>(olb, td);
    k4_xifft <<<NB * NT, 256, 0, stream>>>(td, mig);
    k5_hilbert<<<NB * NXR, 256, 0, stream>>>(mig, logv, cmax);
    k6_bmax  <<<NB, 128, 0, stream>>>(cmax, bmax);
    k7_norm  <<<(NB * NT * NXR) / 256, 256, 0, stream>>>(logv, bmax, out);
}